// GcNetwork_80126909874311
// MI455X (gfx1250) — compile-verified
//
#include <hip/hip_runtime.h>

typedef __attribute__((ext_vector_type(2))) float v2f;
typedef __attribute__((ext_vector_type(8))) float v8f;

constexpr int NN = 50000;   // nodes
constexpr int NE = 800000;  // edges
constexpr int DF = 128;     // feature dim
constexpr int NC = 32;      // classes
constexpr int XPAD = DF + 4; // LDS row pad to dodge bank conflicts

// ---------------------------------------------------------------- utilities
__global__ void k_zero(float* __restrict__ p, int n) {
  int i = blockIdx.x * blockDim.x + threadIdx.x;
  if (i < n) p[i] = 0.0f;
}

__global__ void k_deg(const int* __restrict__ dst, float* __restrict__ deg) {
  int e = blockIdx.x * blockDim.x + threadIdx.x;
  if (e < NE) atomicAdd(&deg[dst[e]], 1.0f);
}

__global__ void k_dinv(float* __restrict__ deg) {
  int i = blockIdx.x * blockDim.x + threadIdx.x;
  if (i < NN) deg[i] = rsqrtf(deg[i] + 1.0f);
}

// ---------------------------------------------------------------- WMMA GEMM
// H = X @ W ; X:[NN,128] row-major, W:[128,128] row-major, H:[NN,128].
// Block = 256 threads = 8 waves. Block owns 16 rows of X; wave w owns output
// columns [16w, 16w+16). Full W (64 KB) staged in LDS (reused by all 8 waves),
// X tile (16x128) staged with +4 padding. 32 x v_wmma_f32_16x16x4_f32 per wave.
// A frag (16x4 f32): lane<16 -> {A[m][k0+0],A[m][k0+1]}, lane>=16 -> K+2.
// B frag (4x16 f32): lane<16 -> {B[k0+0][n],B[k0+1][n]},  lane>=16 -> K+2.
// C/D (16x16 f32):   VGPR i, lane half h -> row i+8h, col lane&15.
__global__ __launch_bounds__(256) void k_gemm128(const float* __restrict__ X,
                                                 const float* __restrict__ W,
                                                 float* __restrict__ H) {
  __shared__ float Ws[DF * DF];      // 64 KB
  __shared__ float Xs[16 * XPAD];    // ~8 KB
  const int tid = threadIdx.x;
  const int m0 = blockIdx.x * 16;    // 50000 % 16 == 0 -> no guards

  // cooperative load of full W (16384 floats, float4 vectorized)
  {
    const float4* Wsrc = (const float4*)W;
    float4* Wdst = (float4*)Ws;
    for (int i = tid; i < DF * DF / 4; i += 256) Wdst[i] = Wsrc[i];
  }
  // load 16x128 X tile into padded LDS
  for (int i = tid; i < 16 * (DF / 4); i += 256) {
    int r = i / (DF / 4), c4 = i % (DF / 4);
    float4 v = *(const float4*)&X[(long long)(m0 + r) * DF + c4 * 4];
    *(float4*)&Xs[r * XPAD + c4 * 4] = v;
  }
  __syncthreads();

  const int wave = tid >> 5;
  const int lane = tid & 31;
  const int half = lane >> 4;
  const int mr = lane & 15;
  const int n0 = wave * 16;

  v8f acc = {};
  for (int k0 = 0; k0 < DF; k0 += 4) {
    const int ka = k0 + 2 * half;
    v2f a = *(const v2f*)&Xs[mr * XPAD + ka];   // 8B LDS read, bank-spread
    v2f b;
    b.x = Ws[ka * DF + n0 + mr];
    b.y = Ws[(ka + 1) * DF + n0 + mr];
    acc = __builtin_amdgcn_wmma_f32_16x16x4_f32(false, a, false, b,
                                                (short)0, acc, false, false);
  }
#pragma unroll
  for (int i = 0; i < 8; ++i)
    H[(long long)(m0 + i + 8 * half) * DF + n0 + mr] = acc[i];
}

// ---------------------------------------------------------------- scatter
// 32 lanes per edge, 4 dims per lane: coalesced 512B gather of h[src],
// 4 global_atomic_add_f32 into agg[dst] (both buffers are L2-resident).
__global__ void k_scatter(const float* __restrict__ h, const int* __restrict__ src,
                          const int* __restrict__ dst, const float* __restrict__ dinv,
                          float* __restrict__ agg) {
  long long t = (long long)blockIdx.x * blockDim.x + threadIdx.x;
  int e = (int)(t >> 5);
  if (e >= NE) return;
  int q = ((int)t & 31) << 2;
  int s = src[e], d = dst[e];
  float nrm = dinv[s] * dinv[d];
  float4 hv = *(const float4*)&h[(long long)s * DF + q];
  float* ap = &agg[(long long)d * DF + q];
  atomicAdd(ap + 0, hv.x * nrm);
  atomicAdd(ap + 1, hv.y * nrm);
  atomicAdd(ap + 2, hv.z * nrm);
  atomicAdd(ap + 3, hv.w * nrm);
}

// ------------------------------------------------- self-loop + bias + ReLU
__global__ void k_finalize(float* __restrict__ agg, const float* __restrict__ h,
                           const float* __restrict__ bias,
                           const float* __restrict__ dinv) {
  long long t = (long long)blockIdx.x * blockDim.x + threadIdx.x;
  if (t >= (long long)NN * DF) return;
  int node = (int)(t >> 7);
  int c = (int)t & (DF - 1);
  float di = dinv[node];
  float v = agg[t] + h[t] * di * di + bias[c];
  agg[t] = fmaxf(v, 0.0f);
}

// ------------------------------------- classifier (128->32) + log_softmax
// one wave32 per node; lane == class; shuffle reductions over the wave.
__global__ void k_classify(const float* __restrict__ h, const float* __restrict__ Wl,
                           const float* __restrict__ bl, float* __restrict__ out) {
  int row = blockIdx.x * (blockDim.x >> 5) + (threadIdx.x >> 5);
  int c = threadIdx.x & 31;
  if (row >= NN) return;
  const float* hr = &h[(long long)row * DF];
  float acc = bl[c];
  for (int k = 0; k < DF; ++k) acc += hr[k] * Wl[k * NC + c];
  float m = acc;
  for (int off = 16; off; off >>= 1) m = fmaxf(m, __shfl_xor(m, off, 32));
  float ex = expf(acc - m);
  float s = ex;
  for (int off = 16; off; off >>= 1) s += __shfl_xor(s, off, 32);
  out[(long long)row * NC + c] = acc - m - logf(s);
}

// ---------------------------------------------------------------- launcher
extern "C" void kernel_launch(void* const* d_in, const int* in_sizes, int n_in,
                              void* d_out, int out_size, void* d_ws, size_t ws_size,
                              hipStream_t stream) {
  const float* x  = (const float*)d_in[0];
  const float* W1 = (const float*)d_in[1];
  const float* b1 = (const float*)d_in[2];
  const float* W2 = (const float*)d_in[3];
  const float* b2 = (const float*)d_in[4];
  const float* W3 = (const float*)d_in[5];
  const float* b3 = (const float*)d_in[6];
  const float* Wl = (const float*)d_in[7];
  const float* bl = (const float*)d_in[8];
  const int*   ei = (const int*)d_in[9];   // [2, NE]
  const int* src = ei;
  const int* dst = ei + NE;
  float* out = (float*)d_out;

  float* bufA = (float*)d_ws;                      // 25.6 MB  (h = X@W)
  float* bufB = bufA + (size_t)NN * DF;            // 25.6 MB  (agg / activations)
  float* dinv = bufB + (size_t)NN * DF;            // 200 KB

  const int T = 256;
  auto nb = [](long long n, int t) { return (int)((n + t - 1) / t); };
  const long long ND = (long long)NN * DF;

  // degree -> dinv (identical for all layers; computed once)
  k_zero<<<nb(NN, T), T, 0, stream>>>(dinv, NN);
  k_deg<<<nb(NE, T), T, 0, stream>>>(dst, dinv);
  k_dinv<<<nb(NN, T), T, 0, stream>>>(dinv);

  const float* in = x;
  const float* Wn[3] = {W1, W2, W3};
  const float* bn[3] = {b1, b2, b3};
  for (int L = 0; L < 3; ++L) {
    k_gemm128<<<NN / 16, 256, 0, stream>>>(in, Wn[L], bufA);
    k_zero<<<nb(ND, T), T, 0, stream>>>(bufB, (int)ND);
    k_scatter<<<nb((long long)NE * 32, T), T, 0, stream>>>(bufA, src, dst, dinv, bufB);
    k_finalize<<<nb(ND, T), T, 0, stream>>>(bufB, bufA, bn[L], dinv);
    in = bufB;
  }
  k_classify<<<nb((long long)NN * 32, T), T, 0, stream>>>(bufB, Wl, bl, out);
}